// KnowformerQKLayer_15951508537885
// MI455X (gfx1250) — compile-verified
//
#include <hip/hip_runtime.h>

#define B_ 4
#define V_ 50000
#define R_ 64
#define D_ 64
#define E_ 500000
#define NROWS (B_ * V_) // 200000

typedef __attribute__((ext_vector_type(2))) float v2f;
typedef __attribute__((ext_vector_type(8))) float v8f;

// ---------------------------------------------------------------------------
// Kernel 0: zero the (V, B*D) accumulator workspace (must be re-zeroed every
// launch; harness does not re-poison between replays).
// ---------------------------------------------------------------------------
__global__ __launch_bounds__(256) void knf_zero(float4* __restrict__ p, int n4) {
  int i = blockIdx.x * 256 + threadIdx.x;
  if (i < n4) p[i] = make_float4(0.f, 0.f, 0.f, 0.f);
}

// ---------------------------------------------------------------------------
// Kernel 1: edge scatter.  msg = zf[etype] * xf[src], segment-summed by dst.
// 64 threads per edge, float4 per thread (256 cols = B*D).  All 32 lanes of a
// wave read the same edge record -> broadcast-coalesced.  Accumulate with
// native gfx1250 global_atomic_add_f32 (no CAS loop).
// agg layout: agg[v][b][d]  (v*256 + b*64 + d)
// ---------------------------------------------------------------------------
__global__ __launch_bounds__(256) void knf_scatter(const float* __restrict__ x,
                                                   const float* __restrict__ z,
                                                   const int* __restrict__ edges,
                                                   float* __restrict__ agg) {
  int tid = blockIdx.x * 256 + threadIdx.x;
  int e = tid >> 6;
  if (e >= E_) return;
  int c = tid & 63;                  // column group: b = c>>4, d0 = (c&15)*4
  int dst = edges[e * 3 + 0];
  int et  = edges[e * 3 + 1];
  int src = edges[e * 3 + 2];
  int b  = c >> 4;
  int d0 = (c & 15) << 2;

  const float4 xv = *(const float4*)(x + (((long)b * V_ + src) << 6) + d0);
  const float4 zv = *(const float4*)(z + (((long)b * R_ + et)  << 6) + d0);

  float* op = agg + ((long)dst << 8) + (b << 6) + d0;
  float m0 = xv.x * zv.x, m1 = xv.y * zv.y, m2 = xv.z * zv.z, m3 = xv.w * zv.w;
  asm volatile("global_atomic_add_f32 %0, %1, off"           :: "v"(op), "v"(m0) : "memory");
  asm volatile("global_atomic_add_f32 %0, %1, off offset:4"  :: "v"(op), "v"(m1) : "memory");
  asm volatile("global_atomic_add_f32 %0, %1, off offset:8"  :: "v"(op), "v"(m2) : "memory");
  asm volatile("global_atomic_add_f32 %0, %1, off offset:12" :: "v"(op), "v"(m3) : "memory");
}

// ---------------------------------------------------------------------------
// Kernel 2: fused  h0 = agg + alpha*x ;  h = relu(h0@W1+b1)@W2+b2 ;
//           out = layernorm(h)*gamma+beta + x
// One wave (32 lanes) owns a 16-row tile; 256-thread block = 8 waves.
// GEMMs use V_WMMA_F32_16X16X4_F32 (fp32 in / fp32 acc, full precision):
//   per layer: 16 K-steps x 4 N-tiles = 64 WMMAs per wave.
// A fragment (16x4 f32): lane L holds row M=L&15, K = 4k + (L>=16 ? 2 : 0)+{0,1}
// B fragment (4x16 f32): lane L holds col N, same K pair -> W stored transposed
//   in LDS so the K pair is one aligned 8-byte LDS load.
// C/D (16x16 f32, 8 VGPRs): VGPR j -> row j (lanes 0-15) / j+8 (lanes 16-31).
// ---------------------------------------------------------------------------
__global__ __launch_bounds__(256) void knf_mlp_ln(const float* __restrict__ x,
                                                  const float* __restrict__ agg,
                                                  const float* __restrict__ W1,
                                                  const float* __restrict__ b1,
                                                  const float* __restrict__ W2,
                                                  const float* __restrict__ b2,
                                                  const float* __restrict__ alpha,
                                                  const float* __restrict__ gamma,
                                                  const float* __restrict__ beta,
                                                  float* __restrict__ out) {
  __shared__ float w1t[64 * 64];        // W1 transposed: w1t[n*64+k] = W1[k][n]
  __shared__ float w2t[64 * 64];
  __shared__ float atile[8][16 * 64];   // per-wave 16x64 staging tile
  __shared__ float musig[8][16][2];     // per-row {mean, rstd}

  const int t = threadIdx.x;
  for (int i = t; i < 4096; i += 256) { // cooperative transposed weight load
    int k = i >> 6, n = i & 63;
    w1t[n * 64 + k] = W1[i];
    w2t[n * 64 + k] = W2[i];
  }
  __syncthreads();

  const int lane = t & 31;
  const int wv   = t >> 5;
  const int row0 = (blockIdx.x * 8 + wv) * 16;
  float* A = atile[wv];

  // ---- Phase 1: stage h0 = agg + alpha*x (row-clamped; stores predicated later)
  const int   c0 = lane * 2;
  const float a0 = alpha[c0], a1 = alpha[c0 + 1];
  for (int r = 0; r < 16; ++r) {
    int row = row0 + r; if (row > NROWS - 1) row = NROWS - 1;
    int bb = row / V_, vv = row - bb * V_;
    const float* xr = x   + (((long)bb * V_ + vv) << 6);
    const float* ar = agg + ((long)vv << 8) + (bb << 6);
    A[r * 64 + c0]     = ar[c0]     + a0 * xr[c0];
    A[r * 64 + c0 + 1] = ar[c0 + 1] + a1 * xr[c0 + 1];
  }
  __syncthreads();

  const int klo  = (lane >> 4) << 1;   // lanes 16-31 hold K+2
  const int arow = lane & 15;          // A row / D column index within N-tile
  const int ncol = lane & 15;
  const int mrow = (lane >> 4) << 3;   // C/D rows: j (lanes<16) or j+8

  v8f acc[4];

  // ---- GEMM 1: relu(h0 @ W1 + b1), bias pre-seeded into C
  #pragma unroll
  for (int nt = 0; nt < 4; ++nt) {
    float bias = b1[nt * 16 + ncol];
    #pragma unroll
    for (int j = 0; j < 8; ++j) acc[nt][j] = bias;
  }
  #pragma unroll
  for (int k = 0; k < 16; ++k) {
    v2f af = *(const v2f*)&A[arow * 64 + 4 * k + klo];
    #pragma unroll
    for (int nt = 0; nt < 4; ++nt) {
      v2f bf = *(const v2f*)&w1t[(nt * 16 + ncol) * 64 + 4 * k + klo];
      acc[nt] = __builtin_amdgcn_wmma_f32_16x16x4_f32(
          false, af, false, bf, (short)0, acc[nt], false, false);
    }
  }
  __syncthreads();
  #pragma unroll
  for (int nt = 0; nt < 4; ++nt) {
    #pragma unroll
    for (int j = 0; j < 8; ++j) {
      float v = acc[nt][j];
      A[(mrow + j) * 64 + nt * 16 + ncol] = v > 0.f ? v : 0.f;  // ReLU
    }
  }
  __syncthreads();

  // ---- GEMM 2: t @ W2 + b2
  #pragma unroll
  for (int nt = 0; nt < 4; ++nt) {
    float bias = b2[nt * 16 + ncol];
    #pragma unroll
    for (int j = 0; j < 8; ++j) acc[nt][j] = bias;
  }
  #pragma unroll
  for (int k = 0; k < 16; ++k) {
    v2f af = *(const v2f*)&A[arow * 64 + 4 * k + klo];
    #pragma unroll
    for (int nt = 0; nt < 4; ++nt) {
      v2f bf = *(const v2f*)&w2t[(nt * 16 + ncol) * 64 + 4 * k + klo];
      acc[nt] = __builtin_amdgcn_wmma_f32_16x16x4_f32(
          false, af, false, bf, (short)0, acc[nt], false, false);
    }
  }
  __syncthreads();
  #pragma unroll
  for (int nt = 0; nt < 4; ++nt) {
    #pragma unroll
    for (int j = 0; j < 8; ++j)
      A[(mrow + j) * 64 + nt * 16 + ncol] = acc[nt][j];
  }
  __syncthreads();

  // ---- LayerNorm stats: one row per lane (lanes 0-15)
  if (lane < 16) {
    float s = 0.f, ss = 0.f;
    #pragma unroll
    for (int c = 0; c < 64; ++c) { float h = A[lane * 64 + c]; s += h; ss += h * h; }
    float mu  = s * (1.f / 64.f);
    float var = ss * (1.f / 64.f) - mu * mu;
    musig[wv][lane][0] = mu;
    musig[wv][lane][1] = rsqrtf(var + 1e-5f);
  }
  __syncthreads();

  // ---- Normalize + affine + residual, store (predicated on valid rows)
  const float g0 = gamma[c0], g1 = gamma[c0 + 1];
  const float be0 = beta[c0], be1 = beta[c0 + 1];
  for (int r = 0; r < 16; ++r) {
    int row = row0 + r;
    if (row < NROWS) {
      int bb = row / V_, vv = row - bb * V_;
      long off = ((long)bb * V_ + vv) << 6;
      const float* xr = x + off;
      float* op = out + off;
      float mu = musig[wv][r][0], rs = musig[wv][r][1];
      op[c0]     = (A[r * 64 + c0]     - mu) * rs * g0 + be0 + xr[c0];
      op[c0 + 1] = (A[r * 64 + c0 + 1] - mu) * rs * g1 + be1 + xr[c0 + 1];
    }
  }
}

// ---------------------------------------------------------------------------
extern "C" void kernel_launch(void* const* d_in, const int* in_sizes, int n_in,
                              void* d_out, int out_size, void* d_ws, size_t ws_size,
                              hipStream_t stream) {
  (void)in_sizes; (void)n_in; (void)out_size; (void)ws_size;
  const float* x     = (const float*)d_in[0];
  const float* z     = (const float*)d_in[1];
  const int*   edges = (const int*)  d_in[2];
  const float* W1    = (const float*)d_in[3];
  const float* b1    = (const float*)d_in[4];
  const float* W2    = (const float*)d_in[5];
  const float* b2    = (const float*)d_in[6];
  const float* alpha = (const float*)d_in[7];
  const float* gamma = (const float*)d_in[8];
  const float* beta  = (const float*)d_in[9];
  float* out = (float*)d_out;
  float* agg = (float*)d_ws;               // V * B * D floats = 51.2 MB scratch

  const int n4 = (V_ * B_ * D_) / 4;       // float4 count for zeroing
  knf_zero<<<(n4 + 255) / 256, 256, 0, stream>>>((float4*)agg, n4);

  knf_scatter<<<(E_ * 64) / 256, 256, 0, stream>>>(x, z, edges, agg);

  const int ntiles = (NROWS + 15) / 16;    // 12500 row-tiles
  const int nblk   = (ntiles + 7) / 8;     // 8 waves (tiles) per block
  knf_mlp_ln<<<nblk, 256, 0, stream>>>(x, agg, W1, b1, W2, b2,
                                       alpha, gamma, beta, out);
}